// DisenConvLayer_69939247448112
// MI455X (gfx1250) — compile-verified
//
#include <hip/hip_runtime.h>
#include <math.h>

typedef __attribute__((ext_vector_type(2))) float v2f;
typedef __attribute__((ext_vector_type(8))) float v8f;

#define IND   256
#define OUTD  64
#define KCAPS 8
#define DD    8
#define ROUTIT 4
#define EPSN  1e-12f

// XOR-lane exchange within groups of 32 via ds_swizzle_b32 (single DS op,
// immediate pattern: offset[14:10]=xor_mask, offset[4:0]=and_mask=0x1F).
// Groups of 8 consecutive lanes stay closed under xor masks 1,2,4.
template <int XMASK>
__device__ __forceinline__ float swz_xor(float v) {
    return __int_as_float(
        __builtin_amdgcn_ds_swizzle(__float_as_int(v), (XMASK << 10) | 0x1F));
}

__device__ __forceinline__ float xor8_sum(float v) {
    v += swz_xor<1>(v);
    v += swz_xor<2>(v);
    v += swz_xor<4>(v);
    return v;
}

__device__ __forceinline__ float xor8_max(float v) {
    v = fmaxf(v, swz_xor<1>(v));
    v = fmaxf(v, swz_xor<2>(v));
    v = fmaxf(v, swz_xor<4>(v));
    return v;
}

// ---------------------------------------------------------------------------
// GEMM: h = l2norm(l2norm(leaky_relu(x @ W + b)))  via V_WMMA_F32_16X16X4_F32
// One wave per 16x16 output tile; 8 waves/block -> 32 rows x 64 cols per block.
// ---------------------------------------------------------------------------
__global__ __launch_bounds__(256) void gemm_h_kernel(const float* __restrict__ x,
                                                     const float* __restrict__ w,
                                                     const float* __restrict__ bias,
                                                     float* __restrict__ h, int n)
{
    const int wave    = threadIdx.x >> 5;
    const int lane    = threadIdx.x & 31;
    const int rowTile = blockIdx.x * 2 + (wave >> 2);
    const int colTile = wave & 3;
    const int rowBase = rowTile * 16;
    if (rowBase >= n) return;              // wave-uniform exit: EXEC stays all-1s for WMMA

    const int half = lane >> 4;            // 0: K pair {0,1}; 1: K pair {2,3} (A) / rows M+8 (D)
    const int l    = lane & 15;
    const int col  = colTile * 16 + l;
    const float* xrow = x + (size_t)(rowBase + l) * IND;

    v8f acc = {};
    for (int k = 0; k < IND; k += 4) {
        const int ka = k + half * 2;
        v2f a, b;
        a.x = xrow[ka];
        a.y = xrow[ka + 1];
        b.x = w[(size_t)ka * OUTD + col];        // W is (256,64) row-major: W[k][col]
        b.y = w[(size_t)(ka + 1) * OUTD + col];
        acc = __builtin_amdgcn_wmma_f32_16x16x4_f32(
            /*neg_a=*/false, a, /*neg_b=*/false, b,
            /*c_mod=*/(short)0, acc, /*reuse_a=*/false, /*reuse_b=*/false);
    }

    const float bc = bias[col];
#pragma unroll
    for (int i = 0; i < 8; ++i) {
        float v = acc[i] + bc;
        v = (v > 0.0f) ? v : 0.01f * v;                  // leaky_relu(0.01)
        // capsule = 8 consecutive cols = 8 consecutive lanes (groups {0-7},{8-15},...)
        float s = xor8_sum(v * v);
        v = v / fmaxf(sqrtf(s), EPSN);                   // first normalize
        float s2 = xor8_sum(v * v);
        v = v / fmaxf(sqrtf(s2), EPSN);                  // second normalize (fidelity)
        const int row = rowBase + half * 8 + i;          // D layout: VGPR i -> M=i / M=i+8
        h[(size_t)row * OUTD + col] = v;
    }
}

// ---------------------------------------------------------------------------
// acc = src (float4 copy)
// ---------------------------------------------------------------------------
__global__ __launch_bounds__(256) void copy4_kernel(const float4* __restrict__ s,
                                                    float4* __restrict__ d, int n4)
{
    const int i = blockIdx.x * blockDim.x + threadIdx.x;
    if (i < n4) d[i] = s[i];
}

// ---------------------------------------------------------------------------
// Edge routing: 8 lanes per edge (one per capsule). Gather z=h[src], ct=c[trg],
// per-capsule dot -> softmax over k (8-lane swizzle cluster) -> atomic scatter.
// h, c, acc all L2-resident (~13 MB each, 192 MB L2); index arrays streamed
// with non-temporal hint so they do not evict the resident capsule buffers.
// ---------------------------------------------------------------------------
__global__ __launch_bounds__(256) void route_kernel(const float* __restrict__ h,
                                                    const float* __restrict__ c,
                                                    const int* __restrict__ trg,
                                                    const int* __restrict__ src,
                                                    float* __restrict__ acc, int m)
{
    const int gid = blockIdx.x * blockDim.x + threadIdx.x;
    const int e   = gid >> 3;
    const int k   = gid & 7;
    if (e >= m) return;          // whole 8-lane cluster exits together (swizzle-safe)

    const int s = __builtin_nontemporal_load(src + e);   // NT: streaming index read
    const int t = __builtin_nontemporal_load(trg + e);

    const float4* zp = (const float4*)(h + (size_t)s * OUTD + k * DD);
    const float4* cp = (const float4*)(c + (size_t)t * OUTD + k * DD);
    const float4 z0 = zp[0], z1 = zp[1];
    const float4 c0 = cp[0], c1 = cp[1];

    float p = z0.x * c0.x + z0.y * c0.y + z0.z * c0.z + z0.w * c0.w
            + z1.x * c1.x + z1.y * c1.y + z1.z * c1.z + z1.w * c1.w;   // TAU = 1

    // softmax over the 8 capsules of this edge (lanes k=0..7 in one cluster)
    const float mx  = xor8_max(p);
    const float ex  = __expf(p - mx);
    const float sum = xor8_sum(ex);
    const float wgt = ex / sum;

    float* ap = acc + (size_t)t * OUTD + k * DD;
    atomicAdd(ap + 0, wgt * z0.x);
    atomicAdd(ap + 1, wgt * z0.y);
    atomicAdd(ap + 2, wgt * z0.z);
    atomicAdd(ap + 3, wgt * z0.w);
    atomicAdd(ap + 4, wgt * z1.x);
    atomicAdd(ap + 5, wgt * z1.y);
    atomicAdd(ap + 6, wgt * z1.z);
    atomicAdd(ap + 7, wgt * z1.w);
}

// ---------------------------------------------------------------------------
// Per-capsule L2 normalize: one thread per (node, capsule) -> 8 contiguous f32.
// ---------------------------------------------------------------------------
__global__ __launch_bounds__(256) void norm_kernel(const float* __restrict__ in,
                                                   float* __restrict__ out, int ncaps)
{
    const int i = blockIdx.x * blockDim.x + threadIdx.x;
    if (i >= ncaps) return;
    const float4* p = (const float4*)(in + (size_t)i * DD);
    float4 a = p[0], b = p[1];
    const float s = a.x * a.x + a.y * a.y + a.z * a.z + a.w * a.w
                  + b.x * b.x + b.y * b.y + b.z * b.z + b.w * b.w;
    const float r = 1.0f / fmaxf(sqrtf(s), EPSN);
    a.x *= r; a.y *= r; a.z *= r; a.w *= r;
    b.x *= r; b.y *= r; b.z *= r; b.w *= r;
    float4* q = (float4*)(out + (size_t)i * DD);
    q[0] = a;
    q[1] = b;
}

extern "C" void kernel_launch(void* const* d_in, const int* in_sizes, int n_in,
                              void* d_out, int out_size, void* d_ws, size_t ws_size,
                              hipStream_t stream)
{
    const float* x    = (const float*)d_in[0];
    const int*   st   = (const int*)d_in[1];       // src_trg (2, M) int
    const float* w    = (const float*)d_in[2];     // (256, 64)
    const float* bias = (const float*)d_in[3];     // (64,)

    const int m = in_sizes[1] / 2;
    const int n = in_sizes[0] / IND;
    const int* trg  = st;          // row 0
    const int* srcI = st + m;      // row 1

    float* h    = (float*)d_ws;                    // (n, 64) fixed message source
    float* c    = h + (size_t)n * OUTD;            // (n, 64) current capsules
    float* accb = c + (size_t)n * OUTD;            // (n, 64) scatter-add target
    float* out  = (float*)d_out;

    // 1) h = l2norm^2(leaky_relu(x @ W + b))   [WMMA f32]
    const int rowTiles  = (n + 15) / 16;
    const int gemmBlks  = (rowTiles + 1) / 2;      // 2 row-tiles per block (8 waves)
    gemm_h_kernel<<<gemmBlks, 256, 0, stream>>>(x, w, bias, h, n);

    // 2) c = h
    const int n4 = (n * OUTD) / 4;
    copy4_kernel<<<(n4 + 255) / 256, 256, 0, stream>>>((const float4*)h, (float4*)c, n4);

    const long long edgeThreads = (long long)m * 8;
    const int edgeBlks = (int)((edgeThreads + 255) / 256);
    const int ncaps    = n * KCAPS;
    const int capBlks  = (ncaps + 255) / 256;

    // 3) routing iterations
    for (int it = 0; it < ROUTIT; ++it) {
        copy4_kernel<<<(n4 + 255) / 256, 256, 0, stream>>>((const float4*)c, (float4*)accb, n4);
        route_kernel<<<edgeBlks, 256, 0, stream>>>(h, c, trg, srcI, accb, m);
        float* dst = (it == ROUTIT - 1) ? out : c;
        norm_kernel<<<capBlks, 256, 0, stream>>>(accb, dst, ncaps);
    }
}